// MultiHeadAttentionQuantum_65429531787393
// MI455X (gfx1250) — compile-verified
//
#include <hip/hip_runtime.h>
#include <math.h>

// MI455X (gfx1250) implementation of "quantum" multi-head attention.
//  - statevector sim fully in registers: wave32 holds 256 complex amps (8/lane)
//  - scores & ctx matmuls via V_WMMA_F32_16X16X4_F32 (D=4 is a perfect K fit)
//  - attn (256 MB, the HBM-bound part) written once, float4-coalesced

typedef __attribute__((ext_vector_type(2))) float v2f;
typedef __attribute__((ext_vector_type(8))) float v8f;

#define B_  32
#define S_  1024
#define E_  8
#define H_  2
#define D_  4
#define NQ  8
#define NL  2
#define LDS_STRIDE 1028   // 1024 + 4 pad: conflict-free WMMA stores & A loads, float4-aligned rows

// ---------------- 8-qubit statevector gates (one wave = one sample) ----------
// index = lane[4:0] : r[2:0]  (wire w lives at bit p = 7-w; p>=3 -> lane bit p-3)

__device__ __forceinline__ void rx_gate(float sx[8], float sy[8], int wire, float c, float s)
{
  const int p = 7 - wire;
  if (p >= 3) {
    const int mask = 1 << (p - 3);
#pragma unroll
    for (int r = 0; r < 8; ++r) {
      float px = __shfl_xor(sx[r], mask, 32);
      float py = __shfl_xor(sy[r], mask, 32);
      float nx = c * sx[r] + s * py;   // new = c*self - i*s*partner
      float ny = c * sy[r] - s * px;
      sx[r] = nx; sy[r] = ny;
    }
  } else {
    const int mm = 1 << p;
    float tx[8], ty[8];
#pragma unroll
    for (int r = 0; r < 8; ++r) { tx[r] = sx[r]; ty[r] = sy[r]; }
#pragma unroll
    for (int r = 0; r < 8; ++r) {
      const int rp = r ^ mm;
      sx[r] = c * tx[r] + s * ty[rp];
      sy[r] = c * ty[r] - s * tx[rp];
    }
  }
}

__device__ __forceinline__ void cnot_ll(float sx[8], float sy[8], int lane, int cb, int tb)
{ // ctrl bit and tgt bit both lane bits
  const int tm = 1 << tb;
  const bool ctrl = (lane >> cb) & 1;
#pragma unroll
  for (int r = 0; r < 8; ++r) {
    float px = __shfl_xor(sx[r], tm, 32);
    float py = __shfl_xor(sy[r], tm, 32);
    if (ctrl) { sx[r] = px; sy[r] = py; }
  }
}

__device__ __forceinline__ void cnot_lr(float sx[8], float sy[8], int lane, int cb, int tp)
{ // ctrl = lane bit, tgt = register bit
  const int tm = 1 << tp;
  if ((lane >> cb) & 1) {
#pragma unroll
    for (int r = 0; r < 8; ++r) {
      if ((r & tm) == 0) {
        const int rp = r | tm;
        float t = sx[r]; sx[r] = sx[rp]; sx[rp] = t;
        t = sy[r]; sy[r] = sy[rp]; sy[rp] = t;
      }
    }
  }
}

__device__ __forceinline__ void cnot_rl(float sx[8], float sy[8], int cp, int tb)
{ // ctrl = register bit, tgt = lane bit
  const int tm = 1 << tb, cm = 1 << cp;
#pragma unroll
  for (int r = 0; r < 8; ++r) {
    float px = __shfl_xor(sx[r], tm, 32);
    float py = __shfl_xor(sy[r], tm, 32);
    if (r & cm) { sx[r] = px; sy[r] = py; }   // compile-time predicate
  }
}

__device__ __forceinline__ void cnot_rr(float sx[8], float sy[8], int cp, int tp)
{ // both register bits
  const int cm = 1 << cp, tm = 1 << tp;
#pragma unroll
  for (int r = 0; r < 8; ++r) {
    if ((r & cm) && !(r & tm)) {
      const int rp = r | tm;
      float t = sx[r]; sx[r] = sx[rp]; sx[rp] = t;
      t = sy[r]; sy[r] = sy[rp]; sy[rp] = t;
    }
  }
}

// AngleEmbedding(RX) + 2x BasicEntangler(RX + CNOT ring), PauliZ expvals.
__global__ __launch_bounds__(128)
void qproj_kernel(const float* __restrict__ in, const float* __restrict__ w,
                  float* __restrict__ out, int nSamples)
{
  const int lane   = threadIdx.x & 31;
  const int sample = blockIdx.x * (blockDim.x >> 5) + (threadIdx.x >> 5);
  if (sample >= nSamples) return;

  float sx[8], sy[8];
#pragma unroll
  for (int r = 0; r < 8; ++r) { sx[r] = 0.f; sy[r] = 0.f; }
  if (lane == 0) sx[0] = 1.f;                 // |00000000>

  const float* x = in + (size_t)sample * NQ;

#pragma unroll
  for (int i = 0; i < NQ; ++i) {              // AngleEmbedding
    float half = 0.5f * x[i];
    rx_gate(sx, sy, i, cosf(half), sinf(half));
  }
#pragma unroll
  for (int l = 0; l < NL; ++l) {              // entangler layers
#pragma unroll
    for (int i = 0; i < NQ; ++i) {
      float half = 0.5f * w[l * NQ + i];
      rx_gate(sx, sy, i, cosf(half), sinf(half));
    }
    cnot_ll(sx, sy, lane, 4, 3);   // CNOT(0,1)
    cnot_ll(sx, sy, lane, 3, 2);   // CNOT(1,2)
    cnot_ll(sx, sy, lane, 2, 1);   // CNOT(2,3)
    cnot_ll(sx, sy, lane, 1, 0);   // CNOT(3,4)
    cnot_lr(sx, sy, lane, 0, 2);   // CNOT(4,5)
    cnot_rr(sx, sy, 2, 1);         // CNOT(5,6)
    cnot_rr(sx, sy, 1, 0);         // CNOT(6,7)
    cnot_rl(sx, sy, 0, 4);         // CNOT(7,0)
  }

  // PauliZ expvals: sum(prob * (1-2*bit_p))
  float prob[8], tot = 0.f;
#pragma unroll
  for (int r = 0; r < 8; ++r) { prob[r] = sx[r]*sx[r] + sy[r]*sy[r]; tot += prob[r]; }
  float e[8];
#pragma unroll
  for (int i = 0; i < 5; ++i)                  // wires 0..4: sign from lane bit 4-i
    e[i] = ((lane >> (4 - i)) & 1) ? -tot : tot;
  e[5] = 0.f; e[6] = 0.f; e[7] = 0.f;
#pragma unroll
  for (int r = 0; r < 8; ++r) {                // wires 5..7: sign from r bits
    e[5] += (r & 4) ? -prob[r] : prob[r];
    e[6] += (r & 2) ? -prob[r] : prob[r];
    e[7] += (r & 1) ? -prob[r] : prob[r];
  }
#pragma unroll
  for (int off = 1; off < 32; off <<= 1) {
#pragma unroll
    for (int i = 0; i < 8; ++i) e[i] += __shfl_xor(e[i], off, 32);
  }
  if (lane == 0) {
    float* o = out + (size_t)sample * NQ;
#pragma unroll
    for (int i = 0; i < 8; ++i) o[i] = e[i];
  }
}

// ---------------- attention: one wave per (b,h,16-row tile) ------------------
__global__ __launch_bounds__(32)
void attn_kernel(const float* __restrict__ qp, const float* __restrict__ kp,
                 const float* __restrict__ vp, float* __restrict__ attn,
                 float* __restrict__ concat)
{
  extern __shared__ float smem[];
  float* sc   = smem;                      // [16][LDS_STRIDE] score tile
  float* vbuf = smem + 16 * LDS_STRIDE;    // [1024][4] V head

  const int lane = threadIdx.x;
  const int tile = blockIdx.x & 63;        // 64 row-tiles of 16
  const int bh   = blockIdx.x >> 6;        // 0..63
  const int h    = bh & 1;
  const int b    = bh >> 1;
  const int hoff = h * D_;

  // stage V head (1024 x 4) into LDS, float4 per row
  for (int s = lane; s < S_; s += 32) {
    float4 vv = *(const float4*)(vp + (size_t)(b * S_ + s) * E_ + hoff);
    *(float4*)(vbuf + s * 4) = vv;
  }
  __syncthreads();

  const int m     = lane & 15;             // N column / A row
  const int k0    = (lane >> 4) * 2;       // K pair per 16x4 f32 A/B layout
  const int rbase = (lane >> 4) * 8;       // D rows: M = rbase + j

  // A = Q tile (16x4): VGPR0/1 hold K={k0,k0+1} of row m
  v2f a = *(const v2f*)(qp + (size_t)(b * S_ + tile * 16 + m) * E_ + hoff + k0);

  // scores = Q Kt / sqrt(D): 64 x WMMA f32 16x16x4 -> LDS
  for (int kt = 0; kt < 64; ++kt) {
    v2f bb = *(const v2f*)(kp + (size_t)(b * S_ + kt * 16 + m) * E_ + hoff + k0);
    v8f c = {};
    c = __builtin_amdgcn_wmma_f32_16x16x4_f32(false, a, false, bb, (short)0, c, false, false);
#pragma unroll
    for (int j = 0; j < 8; ++j)
      sc[(rbase + j) * LDS_STRIDE + kt * 16 + m] = 0.5f * c[j];   // 1/sqrt(4)
  }
  __syncthreads();

  // row softmax: 2 lanes per row, exp stored unnormalized, keep 1/sum
  const int row  = lane >> 1;
  const int cbeg = (lane & 1) * 512;
  float mx = -3.402823e38f;
  for (int ci = 0; ci < 512; ++ci) mx = fmaxf(mx, sc[row * LDS_STRIDE + cbeg + ci]);
  mx = fmaxf(mx, __shfl_xor(mx, 1, 32));
  float sum = 0.f;
  for (int ci = 0; ci < 512; ++ci) {
    float ev = expf(sc[row * LDS_STRIDE + cbeg + ci] - mx);
    sc[row * LDS_STRIDE + cbeg + ci] = ev;
    sum += ev;
  }
  sum += __shfl_xor(sum, 1, 32);
  const float inv = 1.f / sum;
  __syncthreads();

  // attn output tile is 16 consecutive rows of (bh) -> fully contiguous 64KB
  float* ga = attn + ((size_t)bh * S_ + (size_t)tile * 16) * S_;
  for (int it = 0; it < 128; ++it) {
    int fl = (it * 32 + lane) * 4;         // flat float index; row uniform per it
    int rr = fl >> 10;
    float ir = __shfl(inv, rr * 2, 32);
    float4 val = *(const float4*)(sc + rr * LDS_STRIDE + (fl & 1023));
    val.x *= ir; val.y *= ir; val.z *= ir; val.w *= ir;
    *(float4*)(ga + fl) = val;
  }

  // ctx = attn_tile (16x1024) @ V (1024x4): 256 accumulating WMMAs
  const float invm = __shfl(inv, m * 2, 32);
  v8f acc = {};
  for (int kt = 0; kt < 256; ++kt) {
    v2f pa, pb;
    pa.x = sc[m * LDS_STRIDE + kt * 4 + k0]     * invm;
    pa.y = sc[m * LDS_STRIDE + kt * 4 + k0 + 1] * invm;
    float b0 = vbuf[(kt * 4 + k0)     * 4 + (m & 3)];  // safe addr, select below
    float b1 = vbuf[(kt * 4 + k0 + 1) * 4 + (m & 3)];
    pb.x = (m < D_) ? b0 : 0.f;            // B cols 4..15 are zero padding
    pb.y = (m < D_) ? b1 : 0.f;
    acc = __builtin_amdgcn_wmma_f32_16x16x4_f32(false, pa, false, pb, (short)0, acc, false, false);
  }
  if (m < D_) {
#pragma unroll
    for (int j = 0; j < 8; ++j) {
      int rrow = tile * 16 + rbase + j;
      concat[(size_t)(b * S_ + rrow) * E_ + hoff + m] = acc[j];
    }
  }
}

// ---------------------------------------------------------------------------
extern "C" void kernel_launch(void* const* d_in, const int* in_sizes, int n_in,
                              void* d_out, int out_size, void* d_ws, size_t ws_size,
                              hipStream_t stream)
{
  const float* q  = (const float*)d_in[0];
  const float* k  = (const float*)d_in[1];
  const float* v  = (const float*)d_in[2];
  const float* wq = (const float*)d_in[3];
  const float* wk = (const float*)d_in[4];
  const float* wv = (const float*)d_in[5];
  const float* wd = (const float*)d_in[6];

  float* out  = (float*)d_out;                         // (B,S,E) = 262144 floats
  float* attn = out + (size_t)B_ * S_ * E_;            // (B,H,S,S) follows

  float* ws     = (float*)d_ws;                        // 4 x 1MB scratch
  const size_t bse = (size_t)B_ * S_ * E_;
  float* qp     = ws;
  float* kp     = ws + bse;
  float* vp     = ws + 2 * bse;
  float* concat = ws + 3 * bse;

  const int nS = B_ * S_;                              // 32768 samples
  const int blocks = nS / 4;                           // 4 waves / 128-thread block

  qproj_kernel<<<blocks, 128, 0, stream>>>(q, wq, qp, nS);
  qproj_kernel<<<blocks, 128, 0, stream>>>(k, wk, kp, nS);
  qproj_kernel<<<blocks, 128, 0, stream>>>(v, wv, vp, nS);

  const size_t smemBytes = (16 * LDS_STRIDE + S_ * D_) * sizeof(float); // 82176 B
  attn_kernel<<<B_ * H_ * (S_ / 16), 32, smemBytes, stream>>>(qp, kp, vp, attn, concat);

  qproj_kernel<<<blocks, 128, 0, stream>>>(concat, wd, out, nS);
}